// PolyTensorRegression_76450417868857
// MI455X (gfx1250) — compile-verified
//
#include <hip/hip_runtime.h>
#include <math.h>

typedef float v2f __attribute__((ext_vector_type(2)));
typedef float v8f __attribute__((ext_vector_type(8)));

#define DD 4096
#define RR 64
#define BB 8192
#define KC 32                    // K-chunk staged in LDS per buffer
#define NC (DD / KC)             // 128 chunks per pass
#define KP (KC / 2)              // 16 k-pairs per chunk
#define STR 160                  // dwords per k-pair row: 128 data + 32 pad
#define SLICEM (KP * STR)        // 2560 dwords per matrix slice
#define SLICE3 (3 * SLICEM)      // 7680 dwords per staged chunk (30 KB)

// ---------------------------------------------------------------------------
// Staging: LDS layout is [matrix][kpair][r][2] with STR-dword rows, so a
// lane's WMMA B-pair (U[d][r], U[d+1][r]) is one aligned ds_load_b64 at a
// constant offset.  The +32-dword pad puts the two 16-lane halves on
// disjoint bank halves (conflict-free b64 reads).
// ---------------------------------------------------------------------------
__device__ __forceinline__ void stage_load(const float* q0, const float* q1,
                                           const float* q2, int k0, float4 g[6])
{
    const int t = threadIdx.x;
    const float* Q[3] = { q0, q1, q2 };
#pragma unroll
    for (int rnd = 0; rnd < 6; ++rnd) {
        const int idx = rnd * 256 + t;    // float4 index, 0..1535
        const int m   = idx >> 9;         // 512 float4 per matrix
        const int wi  = idx & 511;
        const int d   = wi >> 4;          // 0..31
        const int r0  = (wi & 15) * 4;
        g[rnd] = *reinterpret_cast<const float4*>(Q[m] + (k0 + d) * RR + r0);
    }
}

__device__ __forceinline__ void stage_store(float4 g[6], float* dst)
{
    const int t = threadIdx.x;
#pragma unroll
    for (int rnd = 0; rnd < 6; ++rnd) {
        const int idx = rnd * 256 + t;
        const int m   = idx >> 9;
        const int wi  = idx & 511;
        const int d   = wi >> 4;
        const int r0  = (wi & 15) * 4;
        float* p = dst + m * SLICEM + (d >> 1) * STR + r0 * 2 + (d & 1);
        p[0] = g[rnd].x;                  // direct component stores: no packing
        p[2] = g[rnd].y;
        p[4] = g[rnd].z;
        p[6] = g[rnd].w;
    }
}

// One full K pass: 3 GEMMs (X @ {p0,p1,p2}).  Per K-step: 1 global b64 (A),
// 6 ds_load_2addr_b64 (12 B pairs into distinct regs), then 12
// v_wmma_f32_16x16x4_f32.  Global loads for chunk c+2 prefetched before
// processing chunk c.
template <bool DO_SIN>
__device__ __forceinline__ void run_pass(
    const float* p0, const float* p1, const float* p2,
    const float* xrow, float* smem,
    int half, int l16, v8f acc[3][4], float& sacc)
{
    v8f zero = {};
#pragma unroll
    for (int m = 0; m < 3; ++m)
#pragma unroll
        for (int nt = 0; nt < 4; ++nt)
            acc[m][nt] = zero;

    // Prime both buffers
    {
        float4 g[6];
        stage_load(p0, p1, p2, 0 * KC, g);
        stage_store(g, smem);
        stage_load(p0, p1, p2, 1 * KC, g);
        stage_store(g, smem + SLICE3);
    }

    for (int c = 0; c < NC; ++c) {
        float4 g[6];
        const bool pf = (c + 2 < NC);
        if (pf) stage_load(p0, p1, p2, (c + 2) * KC, g);   // hide L2 latency

        __syncthreads();                  // staged chunk c visible everywhere

        const v2f* bvec = reinterpret_cast<const v2f*>(smem + (c & 1) * SLICE3);
#pragma unroll
        for (int ks = 0; ks < KC / 4; ++ks) {
            const int kl = ks * 4;
            // A fragment: 16x4 f32, M = lane&15, K = c*KC + kl + 2*half + {0,1}
            float2 a2 = *reinterpret_cast<const float2*>(xrow + c * KC + kl);
            v2f a; a.x = a2.x; a.y = a2.y;
            if (DO_SIN) sacc += __sinf(a2.x) + __sinf(a2.y);

            // Phase 1: load all 12 B fragments into distinct registers so
            // the DS loads batch up and waits can be graded.
            const v2f* bp = bvec + (kl / 2 + half) * (STR / 2) + l16;
            v2f b[3][4];
#pragma unroll
            for (int m = 0; m < 3; ++m)
#pragma unroll
                for (int nt = 0; nt < 4; ++nt)
                    b[m][nt] = bp[m * (SLICEM / 2) + nt * 16];

            // Phase 2: 12 back-to-back WMMAs on the matrix pipe.
#pragma unroll
            for (int m = 0; m < 3; ++m)
#pragma unroll
                for (int nt = 0; nt < 4; ++nt)
                    acc[m][nt] = __builtin_amdgcn_wmma_f32_16x16x4_f32(
                        false, a, false, b[m][nt], (short)0, acc[m][nt],
                        false, false);
        }
        __syncthreads();                  // all waves done reading this buf
        if (pf) stage_store(g, smem + (c & 1) * SLICE3);
    }
}

// ---------------------------------------------------------------------------
// Fused CP tensor-regression forward. Two K-passes of 3 GEMMs each keep
// VGPR pressure < 256; X's second pass hits L2 (128 MB of 192 MB L2).
// sin(X) fused into pass 1.
// ---------------------------------------------------------------------------
__global__ __launch_bounds__(256) void poly_main(
    const float* __restrict__ X,
    const float* __restrict__ U1,
    const float* __restrict__ U2,
    const float* __restrict__ U3,
    const float* __restrict__ C,
    const float* __restrict__ beta,
    const float* __restrict__ Csin,
    float* __restrict__ out)
{
    __shared__ float smem[2 * SLICE3];       // 60 KB staging double buffer,
                                             // reused as 8x3x(16x16) term pads

    const int wv   = threadIdx.x >> 5;
    const int lane = threadIdx.x & 31;
    const int half = lane >> 4;              // 0: K%4 in {0,1}, 1: {2,3}
    const int l16  = lane & 15;              // M (A rows) / N (B,C cols)
    const int tile = blockIdx.x * 8 + wv;    // 64 blocks * 8 waves = 512 tiles
    const int row0 = tile * 16;

    const float* xrow = X + (size_t)(row0 + l16) * DD + 2 * half;
    float sacc = 0.0f;

    v8f acc[3][4];

    // Pass 1: U1[0], U2[0], U2[1]  (+ sin accumulation)
    run_pass<true>(U1, U2, U2 + DD * RR, xrow, smem, half, l16, acc, sacc);
    v8f tj0 = acc[0][0] + acc[0][1] + acc[0][2] + acc[0][3];
    v8f tj1 = acc[1][0] * acc[2][0] + acc[1][1] * acc[2][1]
            + acc[1][2] * acc[2][2] + acc[1][3] * acc[2][3];

    // Pass 2: U3[0], U3[1], U3[2]
    run_pass<false>(U3, U3 + DD * RR, U3 + 2 * DD * RR,
                    xrow, smem, half, l16, acc, sacc);
    v8f tj2 = acc[0][0] * acc[1][0] * acc[2][0]
            + acc[0][1] * acc[1][1] * acc[2][1]
            + acc[0][2] * acc[1][2] * acc[2][2]
            + acc[0][3] * acc[1][3] * acc[2][3];

    // full per-row sin sum: the two 16-lane halves cover disjoint K subsets
    sacc += __shfl_xor(sacc, 16);

    // C/D layout: VGPR k, lanes 0-15 -> (M=k, N=lane); lanes 16-31 -> (M=k+8)
    float* ldsw = smem + wv * 768;           // staging buffers are dead now
#pragma unroll
    for (int k = 0; k < 8; ++k) {
        int row = k + 8 * half;
        ldsw[0 * 256 + row * 16 + l16] = tj0[k];
        ldsw[1 * 256 + row * 16 + l16] = tj1[k];
        ldsw[2 * 256 + row * 16 + l16] = tj2[k];
    }
    __syncthreads();

    if (lane < 16) {
        float t0 = 0.f, t1 = 0.f, t2 = 0.f;
#pragma unroll
        for (int n = 0; n < 16; ++n) {
            t0 += ldsw[0 * 256 + l16 * 16 + n];
            t1 += ldsw[1 * 256 + l16 * 16 + n];
            t2 += ldsw[2 * 256 + l16 * 16 + n];
        }
        float res = beta[0] + C[0] * t0 + C[1] * t1 + C[2] * t2 + Csin[0] * sacc;
        out[row0 + l16] = res;
    }
}

// ---------------------------------------------------------------------------
// Regularizer: deterministic two-pass |U| reduction (no float atomics).
// ---------------------------------------------------------------------------
__global__ __launch_bounds__(256) void reg_partial(
    const float* __restrict__ U1,
    const float* __restrict__ U2,
    const float* __restrict__ U3,
    float* __restrict__ ws)
{
    const int seg = blockIdx.x >> 4;   // 0,1,2
    const int blk = blockIdx.x & 15;
    const float* U;
    int n;
    if (seg == 0)      { U = U1; n = 1 * DD * RR; }
    else if (seg == 1) { U = U2; n = 2 * DD * RR; }
    else               { U = U3; n = 3 * DD * RR; }

    float s = 0.f;
    for (int i = blk * 256 + threadIdx.x; i < n; i += 16 * 256)
        s += fabsf(U[i]);

#pragma unroll
    for (int m = 16; m >= 1; m >>= 1)
        s += __shfl_xor(s, m);

    __shared__ float sm[8];
    if ((threadIdx.x & 31) == 0) sm[threadIdx.x >> 5] = s;
    __syncthreads();
    if (threadIdx.x == 0) {
        float b = 0.f;
#pragma unroll
        for (int i = 0; i < 8; ++i) b += sm[i];
        ws[blockIdx.x] = b * (0.01f / (float)n);   // REG_W / U.size
    }
}

__global__ void reg_final(const float* __restrict__ C,
                          const float* __restrict__ Csin,
                          const float* __restrict__ ws,
                          float* __restrict__ out)
{
    if (threadIdx.x == 0) {
        float s = 0.f;
        for (int i = 0; i < 48; ++i) s += ws[i];
        s += 0.05f * (fabsf(C[0]) + fabsf(C[1]) + fabsf(C[2]) + fabsf(Csin[0]));
        out[BB] = s;   // total_reg after the 8192 result elements
    }
}

// ---------------------------------------------------------------------------
extern "C" void kernel_launch(void* const* d_in, const int* in_sizes, int n_in,
                              void* d_out, int out_size, void* d_ws, size_t ws_size,
                              hipStream_t stream)
{
    const float* X    = (const float*)d_in[0];
    const float* U1   = (const float*)d_in[1];
    const float* U2   = (const float*)d_in[2];
    const float* U3   = (const float*)d_in[3];
    const float* C    = (const float*)d_in[4];
    const float* beta = (const float*)d_in[5];
    const float* Csin = (const float*)d_in[6];
    float* out = (float*)d_out;
    float* ws  = (float*)d_ws;

    poly_main<<<64, 256, 0, stream>>>(X, U1, U2, U3, C, beta, Csin, out);
    reg_partial<<<48, 256, 0, stream>>>(U1, U2, U3, ws);
    reg_final<<<1, 32, 0, stream>>>(C, Csin, ws, out);
}